// GradientRefinementModule_37228776522205
// MI455X (gfx1250) — compile-verified
//
#include <hip/hip_runtime.h>

#define GRM_EPS       0.005f
#define GRM_BASE_STEP 0.002f
#define GRM_MAX_ITERS 7

// Linear-interp sample matching jnp: p = clip(q*(L-1), 0, L-1);
// il = floor(p); ir = ceil(p); v = (1-w)*v_l + w*v_r with w = p - il.
// When p is an exact integer, w == 0 so reading row[il+1] (clamped) is
// numerically identical to jnp's ir == il.
__device__ __forceinline__ float grm_sample(const float* __restrict__ row,
                                            float q, int L) {
    float p = q * (float)(L - 1);
    p = fminf(fmaxf(p, 0.0f), (float)(L - 1));
    float fl = floorf(p);
    int il = (int)fl;
    int ir = il + 1;
    if (ir > L - 1) ir = L - 1;
    float w  = p - fl;
    float vl = row[il];
    float vr = row[ir];
    return (1.0f - w) * vl + w * vr;
}

// One Newton-style refinement step (matches _grad_curv + update exactly).
__device__ __forceinline__ float grm_step(const float* __restrict__ row,
                                          float pos, int L) {
    float pc = fminf(fmaxf(pos, GRM_EPS), 1.0f - GRM_EPS);
    float v  = grm_sample(row, pc, L);
    float vl = grm_sample(row, pc - GRM_EPS, L);
    float vr = grm_sample(row, pc + GRM_EPS, L);
    float g = (vr - vl) / (2.0f * GRM_EPS);
    float c = (vr + vl - 2.0f * v) / (GRM_EPS * GRM_EPS);
    c = fminf(fmaxf(c, -1000.0f), 1000.0f);
    float s = -g / (fabsf(c) + 1e-6f);
    s = fminf(fmaxf(s, -0.1f), 0.1f);
    float np = pos + GRM_BASE_STEP * s;
    return fminf(fmaxf(np, 0.0f), 1.0f);
}

// Zero the 7 per-iteration global maxima (must run each call: harness does
// not re-initialize d_ws between replays).
__global__ void grm_init(unsigned* __restrict__ iter_max) {
    if (threadIdx.x < GRM_MAX_ITERS) iter_max[threadIdx.x] = 0u;
}

// Main pass: full 7-step trajectory per element; per-iteration global max of
// the applied |step| via wave32 shuffle tree + one global atomic per block.
__global__ void grm_main(const float* __restrict__ pred,
                         const float* __restrict__ signals,
                         float* __restrict__ out,
                         unsigned* __restrict__ iter_max,
                         int N, int L) {
    __shared__ float warpmax[8];
    const int tid    = blockIdx.x * blockDim.x + threadIdx.x;
    const bool active = (tid < N);
    const int  b     = active ? (tid / 3) : 0;
    const float* row = signals + (size_t)b * (size_t)L;
    float pos = active ? pred[tid] : 0.0f;

    if (active) {
        // Warm the gather window before the address-ALU chain of iteration 1
        // (emits gfx1250 global_prefetch_b8; the 3 sample points stay within
        // a few cachelines for all 7 iterations).
        float pc = fminf(fmaxf(pos, GRM_EPS), 1.0f - GRM_EPS);
        int ic = (int)(pc * (float)(L - 1));
        int iw = (int)(GRM_EPS * (float)(L - 1)) + 4;
        int lo = ic - iw; if (lo < 0) lo = 0;
        int hi = ic + iw; if (hi > L - 1) hi = L - 1;
        __builtin_prefetch(row + lo, 0, 3);
        __builtin_prefetch(row + ic, 0, 3);
        __builtin_prefetch(row + hi, 0, 3);
    }

    const int lane = threadIdx.x & 31;   // wave32
    const int wid  = threadIdx.x >> 5;

    for (int it = 0; it < GRM_MAX_ITERS; ++it) {
        float delta = 0.0f;
        if (active) {
            float np = grm_step(row, pos, L);
            delta = fabsf(np - pos);
            pos = np;
        }
        // wave32 max reduction (5 hops)
        float m = delta;
        #pragma unroll
        for (int off = 16; off > 0; off >>= 1)
            m = fmaxf(m, __shfl_xor(m, off, 32));
        if (lane == 0) warpmax[wid] = m;
        __syncthreads();
        if (wid == 0) {
            float m2 = (lane < (int)(blockDim.x >> 5)) ? warpmax[lane] : 0.0f;
            #pragma unroll
            for (int off = 4; off > 0; off >>= 1)
                m2 = fmaxf(m2, __shfl_xor(m2, off, 32));
            if (lane == 0)
                atomicMax(&iter_max[it], __float_as_uint(m2)); // exact for f32 >= 0
        }
        __syncthreads();  // protect warpmax before next iteration's writes
    }
    if (active) out[tid] = pos;
}

// Finalize: find first iteration k whose global max applied-step < 1e-5
// (scalar `done` in the reference). If none, d_out already holds pos_7.
// Otherwise recompute the k-step truncated trajectory (cold path).
__global__ void grm_finalize(const float* __restrict__ pred,
                             const float* __restrict__ signals,
                             float* __restrict__ out,
                             const unsigned* __restrict__ iter_max,
                             int N, int L) {
    const int tid = blockIdx.x * blockDim.x + threadIdx.x;
    if (tid >= N) return;
    int k = GRM_MAX_ITERS;
    #pragma unroll
    for (int it = 0; it < GRM_MAX_ITERS; ++it) {
        if (__uint_as_float(iter_max[it]) < 1e-5f) { k = it + 1; break; }
    }
    if (k == GRM_MAX_ITERS) return;   // common case: no early global stop
    const float* row = signals + (size_t)(tid / 3) * (size_t)L;
    float pos = pred[tid];
    for (int it = 0; it < k; ++it) pos = grm_step(row, pos, L);
    out[tid] = pos;
}

extern "C" void kernel_launch(void* const* d_in, const int* in_sizes, int n_in,
                              void* d_out, int out_size, void* d_ws, size_t ws_size,
                              hipStream_t stream) {
    const float* pred    = (const float*)d_in[0];   // (8192, 3) f32
    const float* signals = (const float*)d_in[1];   // (8192, 16384) f32
    float* out = (float*)d_out;                     // (8192, 3) f32

    const int N = in_sizes[0];          // B*3
    const int B = N / 3;
    const int L = in_sizes[1] / B;      // 16384

    unsigned* iter_max = (unsigned*)d_ws;  // 7 words of scratch

    grm_init<<<1, 32, 0, stream>>>(iter_max);

    const int threads = 256;            // 8 wave32 waves per block
    const int blocks  = (N + threads - 1) / threads;
    grm_main<<<blocks, threads, 0, stream>>>(pred, signals, out, iter_max, N, L);
    grm_finalize<<<blocks, threads, 0, stream>>>(pred, signals, out, iter_max, N, L);
}